// RobustLoss_50869592653867
// MI455X (gfx1250) — compile-verified
//
#include <hip/hip_runtime.h>

// MI455X / gfx1250, wave32. Pipeline is L2-resident (192MB L2, ~35MB working
// set) and bandwidth-trivial at 23.3 TB/s; median uses deterministic 3-pass
// radix select; 16x16/stride-8 pooling is expressed as f32 WMMA matmuls
// (V_WMMA_F32_16X16X4_F32) with selection matrices per ISA 7.12.2 layouts.
// Stencil inputs for the WMMA A-operand are staged branchlessly through LDS.

typedef __attribute__((ext_vector_type(2))) float v2f;
typedef __attribute__((ext_vector_type(8))) float v8f;

#define H_IMG 2048
#define W_IMG 2048
#define HP_W  256   // pooled width (W/8)

// ws layout (32-bit words):
//   [0..2047]    hist0 (top 11 bits)
//   [2048..4095] hist1 (mid 11 bits)
//   [4096..5119] hist2 (low 10 bits)
//   [5120..5127] sel: 0=b0, 1=prefix22, 2=key, 3=rank-rem, 4=median bits
// byte 32768 : patch[256*256] f32 (256 KB)
// byte 32768+262144 : hp[2048*256] f32 (2 MB)
#define SEL_OFS 5120

__device__ __forceinline__ unsigned fkey(float f) {
  unsigned u = __float_as_uint(f);
  return (u & 0x80000000u) ? ~u : (u | 0x80000000u);  // order-preserving
}

__global__ void k_zero(unsigned* ws) {
  int i = blockIdx.x * blockDim.x + threadIdx.x;
  if (i < 5128) ws[i] = 0u;
}

__global__ void k_hist(const float* __restrict__ R, unsigned n, int pass,
                       unsigned* __restrict__ ws) {
  __shared__ unsigned lh[2048];
  unsigned* hist = ws + (pass == 0 ? 0 : (pass == 1 ? 2048 : 4096));
  const unsigned* sel = ws + SEL_OFS;
  const int nb = (pass == 2) ? 1024 : 2048;
  for (int i = threadIdx.x; i < nb; i += blockDim.x) lh[i] = 0u;
  __syncthreads();
  const unsigned p0 = sel[0], p1 = sel[1];
  const unsigned stride = gridDim.x * blockDim.x;
  for (unsigned i = blockIdx.x * blockDim.x + threadIdx.x; i < n; i += stride) {
    __builtin_prefetch(&R[i + stride], 0, 0);  // global_prefetch_b8
    unsigned k = fkey(R[i]);
    if (pass == 0) {
      atomicAdd(&lh[k >> 21], 1u);
    } else if (pass == 1) {
      if ((k >> 21) == p0) atomicAdd(&lh[(k >> 10) & 2047u], 1u);
    } else {
      if ((k >> 10) == p1) atomicAdd(&lh[k & 1023u], 1u);
    }
  }
  __syncthreads();
  for (int i = threadIdx.x; i < nb; i += blockDim.x) {
    unsigned v = lh[i];
    if (v) atomicAdd(&hist[i], v);
  }
}

__global__ void k_scan(unsigned* ws, int pass, unsigned k_rank) {
  if (threadIdx.x != 0 || blockIdx.x != 0) return;
  unsigned* sel = ws + SEL_OFS;
  const unsigned* hist = ws + (pass == 0 ? 0 : (pass == 1 ? 2048 : 4096));
  const int nb = (pass == 2) ? 1024 : 2048;
  unsigned rank = (pass == 0) ? k_rank : sel[3];
  unsigned cum = 0;
  int b = 0;
  for (; b < nb; ++b) {
    unsigned h = hist[b];
    if (cum + h > rank) break;
    cum += h;
  }
  if (b >= nb) b = nb - 1;  // safety (shouldn't happen)
  if (pass == 0) {
    sel[0] = (unsigned)b; sel[3] = rank - cum;
  } else if (pass == 1) {
    sel[1] = (sel[0] << 11) | (unsigned)b; sel[3] = rank - cum;
  } else {
    unsigned key = (sel[1] << 10) | (unsigned)b;
    sel[2] = key;
    unsigned u = (key & 0x80000000u) ? (key ^ 0x80000000u) : ~key;
    sel[4] = u;  // median as float bits
  }
}

// Horizontal pool via WMMA: hp[r][J] = sum_{dj=0..15} hin[r][clamp(8J-4+dj)]
// where hin = box3x3_zeropad(R - median)*w1 + b1.
// Per 16x16 tile: D = sum_{34 k-chunks} A(hin data 16x4) x B(selection 4x16).
// A staged through LDS; all global loads are unconditional (clamped address
// + predicated zero), so no exec-mask churn in the hot loop.
__global__ void k_hpool(const float* __restrict__ R,
                        const float* __restrict__ w1p, const float* __restrict__ b1p,
                        const unsigned* __restrict__ sel, float* __restrict__ hp) {
  __shared__ float rs[18][138];   // residual strip, zero outside image
  __shared__ float hs[16][137];   // hin strip; stride 137 (9*m mod 64 distinct)
  const float med = __uint_as_float(sel[4]);
  const float w1_9 = w1p[0] * (1.f / 9.f), b1 = b1p[0];
  const int TJ = blockIdx.x;        // 0..15  (J tiles of 16)
  const int r0 = blockIdx.y * 16;   // 0..2032 (row tiles)
  const int lane = threadIdx.x;     // 0..31
  const int cbase = TJ * 128 - 4;

  // Stage 1: residual strip rows r0-1..r0+16, cols cbase-1..cbase+136.
  for (int idx = lane; idx < 18 * 138; idx += 32) {
    const int rr = idx / 138, jj = idx - rr * 138;
    const int gi = r0 - 1 + rr;
    const int gj = cbase - 1 + jj;
    const int gic = min(max(gi, 0), H_IMG - 1);
    const int gjc = min(max(gj, 0), W_IMG - 1);
    const float v = R[gic * W_IMG + gjc];           // always valid address
    const bool ok = (gi >= 0) & (gi < H_IMG) & (gj >= 0) & (gj < W_IMG);
    rs[rr][jj] = ok ? v : 0.f;                      // zero-pad via cndmask
  }
  __syncthreads();

  // Stage 2: hin strip, entry ka holds hin(r0+m, clamp(cbase+ka)).
  for (int idx = lane; idx < 16 * 136; idx += 32) {
    const int m = idx / 136, ka = idx - m * 136;
    const int r = r0 + m;
    const int cc = min(max(cbase + ka, 0), W_IMG - 1);
    const int j = cc - cbase + 1;                   // center col in rs
    float s = rs[m][j - 1]     + rs[m][j]     + rs[m][j + 1]
            + rs[m + 1][j - 1] + rs[m + 1][j] + rs[m + 1][j + 1]
            + rs[m + 2][j - 1] + rs[m + 2][j] + rs[m + 2][j + 1];
    const int cnt = ((r > 0) + 1 + (r < H_IMG - 1)) *
                    ((cc > 0) + 1 + (cc < W_IMG - 1));
    hs[m][ka] = (s - (float)cnt * med) * w1_9 + b1;
  }
  __syncthreads();

  // Stage 3: 34 WMMA accumulations, A from LDS, B = 0/1 window selection.
  const int m  = lane & 15;   // A: M index, B: N index
  const int kg = lane >> 4;   // K group (K = 4c + 2*kg + v)
  v8f acc = {};
  for (int c = 0; c < 34; ++c) {
    const int ka0 = 4 * c + 2 * kg, ka1 = ka0 + 1;
    v2f a;  a.x = hs[m][ka0];
            a.y = hs[m][ka1];
    v2f bm; bm.x = (ka0 >= 8 * m && ka0 <= 8 * m + 15) ? 1.f : 0.f;
            bm.y = (ka1 >= 8 * m && ka1 <= 8 * m + 15) ? 1.f : 0.f;
    acc = __builtin_amdgcn_wmma_f32_16x16x4_f32(false, a, false, bm,
                                                (short)0, acc, false, false);
  }
  const int J = TJ * 16 + m;
#pragma unroll
  for (int v = 0; v < 8; ++v)  // D layout: lane m holds N=m, rows M=v+8*kg
    hp[(r0 + v + 8 * kg) * HP_W + J] = acc[v];
}

// Vertical pool via WMMA: patch[I][J] = (w2/256)*sum_{di} hp[clamp(8I-4+di)][J] + b2
__global__ void k_vpool(const float* __restrict__ hp,
                        const float* __restrict__ w2p, const float* __restrict__ b2p,
                        float* __restrict__ patch) {
  const float w2s = w2p[0] * (1.f / 256.f), b2 = b2p[0];
  const int TJ = blockIdx.x, TI = blockIdx.y;  // 16x16 tiles of patch
  const int lane = threadIdx.x;
  const int m  = lane & 15;
  const int kg = lane >> 4;
  const int rbase = TI * 128 - 4;
  const int J = TJ * 16 + m;
  v8f acc = {};
  for (int c = 0; c < 34; ++c) {
    const int ka0 = 4 * c + 2 * kg, ka1 = ka0 + 1;
    const int rr0 = min(max(rbase + ka0, 0), H_IMG - 1);
    const int rr1 = min(max(rbase + ka1, 0), H_IMG - 1);
    v2f a;  a.x = (ka0 >= 8 * m && ka0 <= 8 * m + 15) ? w2s : 0.f;  // sel*w2/256
            a.y = (ka1 >= 8 * m && ka1 <= 8 * m + 15) ? w2s : 0.f;
    v2f bm; bm.x = hp[rr0 * HP_W + J];
            bm.y = hp[rr1 * HP_W + J];
    acc = __builtin_amdgcn_wmma_f32_16x16x4_f32(false, a, false, bm,
                                                (short)0, acc, false, false);
  }
#pragma unroll
  for (int v = 0; v < 8; ++v)
    patch[(TI * 16 + v + 8 * kg) * HP_W + J] = acc[v] + b2;
}

// Final: out = sigmoid((patch[i/8][j/8] + hin(i,j) + (R-med))*w3 + b3)
__global__ void k_final(const float* __restrict__ R, const float* __restrict__ patch,
                        const unsigned* __restrict__ sel,
                        const float* __restrict__ w1p, const float* __restrict__ b1p,
                        const float* __restrict__ w3p, const float* __restrict__ b3p,
                        float* __restrict__ out) {
  __shared__ float t[18][18];
  const float med = __uint_as_float(sel[4]);
  const float w1_9 = w1p[0] * (1.f / 9.f), b1 = b1p[0];
  const float w3 = w3p[0], b3 = b3p[0];
  const int bi = blockIdx.y * 16, bj = blockIdx.x * 16;
  const int tx = threadIdx.x & 15, ty = threadIdx.x >> 4;
  for (int idx = threadIdx.x; idx < 18 * 18; idx += 256) {
    const int rr = idx / 18, cc = idx - rr * 18;
    const int gi = bi + rr - 1, gj = bj + cc - 1;
    const int gic = min(max(gi, 0), H_IMG - 1);
    const int gjc = min(max(gj, 0), W_IMG - 1);
    const float v = R[gic * W_IMG + gjc];           // always valid address
    const bool ok = (gi >= 0) & (gi < H_IMG) & (gj >= 0) & (gj < W_IMG);
    t[rr][cc] = ok ? v : 0.f;
  }
  __syncthreads();
  const int i = bi + ty, j = bj + tx;
  float s = t[ty][tx]     + t[ty][tx + 1]     + t[ty][tx + 2]
          + t[ty + 1][tx] + t[ty + 1][tx + 1] + t[ty + 1][tx + 2]
          + t[ty + 2][tx] + t[ty + 2][tx + 1] + t[ty + 2][tx + 2];
  const int cnt = ((i > 0) + 1 + (i < H_IMG - 1)) * ((j > 0) + 1 + (j < W_IMG - 1));
  const float hin = (s - (float)cnt * med) * w1_9 + b1;
  const float inl = t[ty + 1][tx + 1] - med;
  const float p = patch[(i >> 3) * HP_W + (j >> 3)];
  const float x = (p + hin + inl) * w3 + b3;
  out[i * W_IMG + j] = 1.f / (1.f + __expf(-x));
}

extern "C" void kernel_launch(void* const* d_in, const int* in_sizes, int n_in,
                              void* d_out, int out_size, void* d_ws, size_t ws_size,
                              hipStream_t stream) {
  const float* R  = (const float*)d_in[0];
  const float* w1 = (const float*)d_in[1];
  const float* b1 = (const float*)d_in[2];
  const float* w2 = (const float*)d_in[3];
  const float* b2 = (const float*)d_in[4];
  const float* w3 = (const float*)d_in[5];
  const float* b3 = (const float*)d_in[6];
  float* out = (float*)d_out;

  unsigned n = (unsigned)in_sizes[0];       // 2048*2048
  unsigned krank = (n - 1) / 2;             // lower median, 0-indexed

  unsigned* ws   = (unsigned*)d_ws;
  float* patch   = (float*)((char*)d_ws + 32768);
  float* hp      = (float*)((char*)d_ws + 32768 + HP_W * HP_W * 4);
  const unsigned* sel = ws + SEL_OFS;

  k_zero<<<21, 256, 0, stream>>>(ws);
  for (int pass = 0; pass < 3; ++pass) {
    k_hist<<<512, 256, 0, stream>>>(R, n, pass, ws);
    k_scan<<<1, 32, 0, stream>>>(ws, pass, krank);
  }
  k_hpool<<<dim3(16, 128), 32, 0, stream>>>(R, w1, b1, sel, hp);
  k_vpool<<<dim3(16, 16), 32, 0, stream>>>(hp, w2, b2, patch);
  k_final<<<dim3(128, 128), 256, 0, stream>>>(R, patch, sel, w1, b1, w3, b3, out);
}